// EdgePoolingLayer_18451179504186
// MI455X (gfx1250) — compile-verified
//
#include <hip/hip_runtime.h>
#include <cmath>

typedef __attribute__((ext_vector_type(2))) float v2f;
typedef __attribute__((ext_vector_type(8))) float v8f;

#define BATCH 4
#define CDIM  256
#define NPTS  2048
#define KNN   32
#define KEEP  1024

// ---------------------------------------------------------------------------
// Kernel A: per-point scalars  sq[n]=|x_n|^2,  dw[n]=w_d.x_n,  cw[n]=w_c.x_n
// ---------------------------------------------------------------------------
__global__ __launch_bounds__(256)
void precompute_kernel(const float* __restrict__ x, const float* __restrict__ w,
                       float* __restrict__ sq, float* __restrict__ dw,
                       float* __restrict__ cw) {
    int g = blockIdx.x * 256 + threadIdx.x;      // 0 .. B*N-1
    int b = g >> 11;                             // / NPTS
    int n = g & (NPTS - 1);
    const float* xb = x + (size_t)b * CDIM * NPTS;
    float s = 0.f, dv = 0.f, cv = 0.f;
    for (int c = 0; c < CDIM; ++c) {
        float v = xb[c * NPTS + n];              // coalesced over n
        s += v * v;
        cv += w[c] * v;                          // w_c = w[:C]
        dv += w[CDIM + c] * v;                   // w_d = w[C:]
    }
    sq[g] = s; dw[g] = dv; cw[g] = cv;
}

// ---------------------------------------------------------------------------
// Kernel B: f32 WMMA Gram tiles -> distance keys in LDS -> 32-NN extraction
// One workgroup = batch b, 16 query rows. 16 waves (512 threads).
// ---------------------------------------------------------------------------
__global__ __launch_bounds__(512)
void knn_score_kernel(const float* __restrict__ x,
                      const float* __restrict__ sq,
                      const float* __restrict__ dw,
                      const float* __restrict__ cw,
                      const float* __restrict__ bias_p,
                      float* __restrict__ scores) {
    __shared__ float As[CDIM * 16];       // A block: x[b, c, n0..n0+15]  (16 KB)
    __shared__ float distS[16 * NPTS];    // keys for 16 rows x 2048 cols (128 KB)

    const int blk  = blockIdx.x;          // 0 .. B*(N/16)-1
    const int b    = blk >> 7;            // / (N/16)
    const int n0   = (blk & 127) << 4;
    const int tid  = threadIdx.x;
    const int lane = tid & 31;
    const int wid  = tid >> 5;            // 0..15
    const float* xb = x + (size_t)b * CDIM * NPTS;

    // ---- stage 1: stage A block into LDS, layout As[c*16 + j] ----
    for (int i = tid; i < CDIM * 16; i += 512) {
        int c = i >> 4, j = i & 15;
        As[i] = xb[c * NPTS + n0 + j];
    }
    __syncthreads();

    // ---- stage 2: WMMA f32 16x16x4 over C, keys into distS ----
    // A frag layout: lanes 0-15 hold M=lane, (K=0,K=1); lanes 16-31 hold (K=2,K=3)
    // B frag layout: lanes 0-15 hold N=lane, (K=0,K=1); lanes 16-31 hold (K=2,K=3)
    const int l15 = lane & 15;
    const int hi  = lane >> 4;            // 0 or 1
    for (int t = wid; t < NPTS / 16; t += 16) {
        int m = t * 16 + l15;             // this lane's column
        float sqm = sq[b * NPTS + m];
        v8f acc = {0.f, 0.f, 0.f, 0.f, 0.f, 0.f, 0.f, 0.f};
        for (int c = 0; c < CDIM; c += 4) {
            int ck = c + (hi << 1);
            v2f av, bv;
            av.x = As[ck * 16 + l15];
            av.y = As[(ck + 1) * 16 + l15];
            bv.x = xb[ck * NPTS + m];
            bv.y = xb[(ck + 1) * NPTS + m];
            acc = __builtin_amdgcn_wmma_f32_16x16x4_f32(
                false, av, false, bv, (short)0, acc, false, false);
        }
        // D layout: VGPR v -> row v (lanes 0-15) / row v+8 (lanes 16-31)
#pragma unroll
        for (int v = 0; v < 8; ++v) {
            int row = v + (hi << 3);
            distS[row * NPTS + m] = sqm - 2.0f * acc[v];   // rank key (sq[n] dropped)
        }
    }
    __syncthreads();

    // ---- stage 3: wave `wid` extracts 32 smallest keys of row `wid` ----
    float* rowp = distS + wid * NPTS;
    const float FINF = __builtin_inff();
    const float* dwb = dw + b * NPTS;
    float maxd = -FINF;
    const int base = lane * (NPTS / 32);
    for (int it = 0; it < KNN; ++it) {
        float best = FINF; int bidx = -1;
        for (int mm = 0; mm < NPTS / 32; ++mm) {
            float vv = rowp[base + mm];
            if (vv < best) { best = vv; bidx = base + mm; }   // ties -> lower index
        }
        // wave32 argmin reduction, lower index wins on ties
        for (int off = 16; off > 0; off >>= 1) {
            float ob = __shfl_xor(best, off, 32);
            int   oi = __shfl_xor(bidx, off, 32);
            if (ob < best || (ob == best && (unsigned)oi < (unsigned)bidx)) {
                best = ob; bidx = oi;
            }
        }
        if (lane == 0) rowp[bidx] = FINF;     // remove winner (same-wave DS is in-order)
        maxd = fmaxf(maxd, dwb[bidx]);        // bidx uniform across lanes
    }
    if (lane == 0) {
        int n = n0 + wid;
        float s = cw[b * NPTS + n] - dwb[n] + bias_p[0] + maxd;
        scores[b * NPTS + n] = tanhf(s);
    }
}

// ---------------------------------------------------------------------------
// Kernel C: per-batch bitonic sort of 2048 scores (desc, ties -> lower index)
// ---------------------------------------------------------------------------
__global__ __launch_bounds__(1024)
void topk_sort_kernel(const float* __restrict__ scores,
                      float* __restrict__ tsc, int* __restrict__ tix) {
    __shared__ float ky[NPTS];
    __shared__ int   id[NPTS];
    const int b = blockIdx.x;
    const int t = threadIdx.x;            // 0..1023
    for (int i = t; i < NPTS; i += 1024) { ky[i] = scores[b * NPTS + i]; id[i] = i; }
    __syncthreads();
    for (int size = 2; size <= NPTS; size <<= 1) {
        for (int stride = size >> 1; stride > 0; stride >>= 1) {
            int i = ((t & ~(stride - 1)) << 1) | (t & (stride - 1));
            int j = i | stride;
            bool asc = ((i & size) == 0);     // comp-ascending block
            float ki = ky[i], kj = ky[j];
            int   ii = id[i], ij = id[j];
            // "less" in comp-space == better (higher score, or tie w/ lower idx)
            bool iLess = (ki > kj) || (ki == kj && ii < ij);
            bool doswap = asc ? !iLess : iLess;
            if (doswap) { ky[i] = kj; ky[j] = ki; id[i] = ij; id[j] = ii; }
            __syncthreads();
        }
    }
    if (t < KEEP) {                       // first 1024 = top, already sorted
        tsc[b * KEEP + t] = ky[t];
        tix[b * KEEP + t] = id[t];
    }
}

// ---------------------------------------------------------------------------
// Kernel D: out[b,c,j] = x[b,c,idx[j]] * score[j]
// ---------------------------------------------------------------------------
__global__ __launch_bounds__(1024)
void gather_scale_kernel(const float* __restrict__ x,
                         const float* __restrict__ tsc,
                         const int* __restrict__ tix,
                         float* __restrict__ out) {
    int bc = blockIdx.x;                  // 0 .. B*C-1
    int b  = bc >> 8;                     // / CDIM
    int j  = threadIdx.x;                 // 0..1023
    float s = tsc[b * KEEP + j];
    int   n = tix[b * KEEP + j];
    out[(size_t)bc * KEEP + j] = x[(size_t)bc * NPTS + n] * s;
}

// ---------------------------------------------------------------------------
extern "C" void kernel_launch(void* const* d_in, const int* in_sizes, int n_in,
                              void* d_out, int out_size, void* d_ws, size_t ws_size,
                              hipStream_t stream) {
    const float* x    = (const float*)d_in[0];   // (4,256,2048) f32
    const float* w    = (const float*)d_in[1];   // (512,) f32
    const float* bias = (const float*)d_in[2];   // (1,) f32
    float* out = (float*)d_out;                  // (4,256,1024) f32

    float* ws     = (float*)d_ws;
    float* sq     = ws;                          // B*N
    float* dw     = ws + BATCH * NPTS;           // B*N
    float* cw     = ws + 2 * BATCH * NPTS;       // B*N
    float* scores = ws + 3 * BATCH * NPTS;       // B*N
    float* tsc    = ws + 4 * BATCH * NPTS;       // B*KEEP
    int*   tix    = (int*)(ws + 4 * BATCH * NPTS + BATCH * KEEP);

    precompute_kernel<<<(BATCH * NPTS) / 256, 256, 0, stream>>>(x, w, sq, dw, cw);
    knn_score_kernel<<<BATCH * (NPTS / 16), 512, 0, stream>>>(x, sq, dw, cw, bias, scores);
    topk_sort_kernel<<<BATCH, 1024, 0, stream>>>(scores, tsc, tix);
    gather_scale_kernel<<<BATCH * CDIM, 1024, 0, stream>>>(x, tsc, tix, out);
    (void)in_sizes; (void)n_in; (void)out_size; (void)ws_size;
}